// alphaLSTMNetwork_61718680044352
// MI455X (gfx1250) — compile-verified
//
#include <hip/hip_runtime.h>
#include <hip/hip_bf16.h>
#include <cstdint>

// ---------------------------------------------------------------------------
// alpha-LSTM elevator network, MI455X (gfx1250) persistent-kernel design.
//
// Latency-bound sequential RNN: one workgroup (8 wave32 waves) loops over all
// 2048 time steps. All batch-16 matmuls use V_WMMA_F32_16X16X4_F32 (M=16 ==
// number of elevators). Weights are pre-packed into WMMA B-fragment order in
// d_ws for coalesced L2-resident streaming. The serial 48-micro-step comm
// scan runs as split-K matvecs across all 256 lanes with LDS-held weights.
// Feature tiles are double-buffered in LDS and prefetched one step ahead with
// the Tensor Data Mover (tensor_load_to_lds + s_wait_tensorcnt), taking the
// global->LDS latency off the 2048-iteration serial critical path.
// ---------------------------------------------------------------------------

typedef float v2f __attribute__((ext_vector_type(2)));
typedef float v4f __attribute__((ext_vector_type(4)));
typedef float v8f __attribute__((ext_vector_type(8)));
typedef unsigned int u32x4 __attribute__((ext_vector_type(4)));
typedef int i32x4 __attribute__((ext_vector_type(4)));
typedef int i32x8 __attribute__((ext_vector_type(8)));

#if defined(__has_builtin)
#if __has_builtin(__builtin_amdgcn_tensor_load_to_lds) && \
    __has_builtin(__builtin_amdgcn_s_wait_tensorcnt)
#define HAVE_TDM 1
#endif
#endif

#define E_   16
#define F_   128
#define PRE_ 64
#define CI_  64
#define CH_  32
#define OUT_ 32
#define NF_  64

// ----- packed B-fragment layout in d_ws (floats) ---------------------------
// fragment = 64 floats: lane*2+{0,1} -> W[n0+(lane&15)][4*ks+2*(lane>>4)+j]
constexpr int P0X  = 0;                          // pre_Wih0: 16 tiles x 32 ksteps
constexpr int P0H  = P0X  + 16*32*64;            // pre_Whh0: 16 x 16
constexpr int P1X  = P0H  + 16*16*64;            // pre_Wih1: 16 x 16
constexpr int P1H  = P1X  + 16*16*64;            // pre_Whh1: 16 x 16
constexpr int PFC  = P1H  + 16*16*64;            // fc_W:      4 x 16
constexpr int POW  = PFC  +  4*16*64;            // out_Wih:   8 x 48
constexpr int POH  = POW  +  8*48*64;            // out_Whh:   8 x  8
constexpr int PTAR = POH  +  8* 8*64;            // tar_W:     4 x  8
constexpr int PACK_TOTAL = PTAR + 4*8*64;        // = 116736 floats

// ----- LDS layout (floats), padded strides to dodge 64-bank conflicts ------
constexpr int LDF = 132, LDH = 66, LDX = 66, LDOI = 196, LDOH = 34, LDG = 256;
constexpr int O_FEAT0 = 0;                       // feature double buffer
constexpr int O_FEAT1 = O_FEAT0 + E_*LDF;
constexpr int O_HP0  = O_FEAT1 + E_*LDF;
constexpr int O_CP0  = O_HP0  + E_*LDH;
constexpr int O_HP1  = O_CP0  + E_*LDH;
constexpr int O_CP1  = O_HP1  + E_*LDH;
constexpr int O_X    = O_CP1  + E_*LDH;
constexpr int O_OI   = O_X    + E_*LDX;
constexpr int O_OH   = O_OI   + E_*LDOI;
constexpr int O_OC   = O_OH   + E_*LDOH;
constexpr int O_G    = O_OC   + E_*LDOH;
constexpr int O_GRP  = O_G    + E_*LDG;
constexpr int O_CH0  = O_GRP  + 64;
constexpr int O_CH1  = O_CH0  + 32;
constexpr int O_PA   = O_CH1  + 32;
constexpr int O_PB   = O_PA   + 128;
constexpr int O_DIR  = O_PB   + 128;
constexpr int O_W0   = O_DIR  + 48;              // comm Wih0 128x64 pad65
constexpr int O_WH0  = O_W0   + 128*65;          // comm Whh0 128x32 pad33
constexpr int O_W1   = O_WH0  + 128*33;          // comm Wih1 128x32 pad33
constexpr int O_WH1  = O_W1   + 128*33;          // comm Whh1 128x32 pad33
constexpr int O_B0   = O_WH1  + 128*33;          // comm bias0 (bih+bhh) 128
constexpr int O_B1   = O_B0   + 128;             // comm bias1 128
constexpr int LDS_FLOATS = O_B1 + 128;           // ~155 KB (WGP has 320 KB)

// ---------------------------------------------------------------------------
__device__ __forceinline__ float sigm(float x) { return 1.0f / (1.0f + __expf(-x)); }
__device__ __forceinline__ float tanh_f(float x) {
    float a = fabsf(x);
    float e = __expf(-2.0f * a);
    float t = (1.0f - e) / (1.0f + e);
    return copysignf(t, x);
}

__device__ __forceinline__ v8f bias8(const float* b1, const float* b2, int n0, int lane) {
    int n = n0 + (lane & 15);
    float v = b1[n] + (b2 ? b2[n] : 0.0f);
    v8f a;
#pragma unroll
    for (int r = 0; r < 8; ++r) a[r] = v;
    return a;
}

__device__ __forceinline__ void store_frag(float* G, int n0, int lane, v8f c) {
    int n  = n0 + (lane & 15);
    int mh = (lane >> 4) * 8;
#pragma unroll
    for (int r = 0; r < 8; ++r) G[(mh + r) * LDG + n] = c[r];
}

// D[16,16] += A[16,4k] * W^T tile, KS k-steps of 4, via V_WMMA_F32_16X16X4_F32.
// A layout: lanes 0-15 rows M, VGPR0=K{0|2}, VGPR1=K{1|3}.  B analogous (KxN).
template <int KS>
__device__ __forceinline__ v8f mm_frag(v8f acc, const float* __restrict__ As, int ldA,
                                       const float* __restrict__ pkb,
                                       const float* __restrict__ W, int ldW,
                                       int n0, int lane) {
    const int rw = lane & 15, kh = lane >> 4;
    const float* ap = As + rw * ldA + 2 * kh;
    if (pkb) {                                   // packed: coalesced 256B frags
        const float* bp = pkb + 2 * lane;
#pragma unroll
        for (int ks = 0; ks < KS; ++ks) {
            v2f a = *(const v2f*)(ap + 4 * ks);
            v2f b = *(const v2f*)(bp + 64 * ks);
            acc = __builtin_amdgcn_wmma_f32_16x16x4_f32(false, a, false, b,
                                                        (short)0, acc, false, false);
        }
    } else {                                     // fallback: strided from weights
        const float* bp = W + (size_t)(n0 + rw) * ldW + 2 * kh;
#pragma unroll
        for (int ks = 0; ks < KS; ++ks) {
            v2f a = *(const v2f*)(ap + 4 * ks);
            v2f b = *(const v2f*)(bp + 4 * ks);
            acc = __builtin_amdgcn_wmma_f32_16x16x4_f32(false, a, false, b,
                                                        (short)0, acc, false, false);
        }
    }
    return acc;
}

#ifdef HAVE_TDM
// ---------------------------------------------------------------------------
// TDM async prefetch of one feature tile: 2-D tensor 128 (dim0) x 16 rows,
// data_size=4B. LDS padding: pad_interval code 6 (=128 DWORDs) + pad_amount
// code 3 (=4 DWORDs) reproduces the LDS row stride of 132 floats.
// D# bitfields per CDNA5 ISA.
// clang-23 toolchain builtin arity: 6 args
//   (u32x4 g0, i32x8 g1, i32x4 g2, i32x4 g3, i32x8 extra, i32 cpol)
__device__ __forceinline__ void tdm_load_feat(const float* gsrc, unsigned ldsByteOff) {
    unsigned long long ga = (unsigned long long)(uintptr_t)gsrc;
    u32x4 g0;
    g0[0] = 1u;                                               // count=1, user D#
    g0[1] = ldsByteOff;                                       // lds_addr (bytes)
    g0[2] = (unsigned)(ga & 0xFFFFFFFFull);                   // global_addr[31:0]
    g0[3] = (unsigned)((ga >> 32) & 0x01FFFFFFull) | (2u << 30); // addr[56:32], type=2
    i32x8 g1;
    g1[0] = (2 << 16) | (1 << 20) | (6 << 22) | (3 << 25);    // dsz=4B, pad on, 128dw/+4dw
    g1[1] = (int)((128u & 0xFFFFu) << 16);                    // tensor_dim0[15:0]
    g1[2] = (int)((16u & 0xFFFFu) << 16);                     // dim0 hi, tensor_dim1[15:0]
    g1[3] = (int)(128u << 16);                                // dim1 hi, tile_dim0=128
    g1[4] = 16;                                               // tile_dim1=16, tile_dim2=0
    g1[5] = 128;                                              // tensor_dim0_stride lo
    g1[6] = 0;
    g1[7] = 0;
    i32x4 z4 = {0, 0, 0, 0};
    i32x8 z8 = {0, 0, 0, 0, 0, 0, 0, 0};
    __builtin_amdgcn_tensor_load_to_lds(g0, g1, z4, z4, z8, 0);
}
#endif

// ---------------------------------------------------------------------------
// One-time weight repack into WMMA B-fragment order.
__global__ void pack_kernel(float* __restrict__ ws,
                            const float* w0x, const float* w0h,
                            const float* w1x, const float* w1h,
                            const float* wfc, const float* wow,
                            const float* woh, const float* wtar) {
    int i = blockIdx.x * 256 + threadIdx.x;
    if (i >= PACK_TOTAL) return;
    const float* W; int base, ks, ldW;
    if      (i < P0H ) { base = P0X;  ks = 32; ldW = 128; W = w0x; }
    else if (i < P1X ) { base = P0H;  ks = 16; ldW = 64;  W = w0h; }
    else if (i < P1H ) { base = P1X;  ks = 16; ldW = 64;  W = w1x; }
    else if (i < PFC ) { base = P1H;  ks = 16; ldW = 64;  W = w1h; }
    else if (i < POW ) { base = PFC;  ks = 16; ldW = 64;  W = wfc; }
    else if (i < POH ) { base = POW;  ks = 48; ldW = 192; W = wow; }
    else if (i < PTAR) { base = POH;  ks = 8;  ldW = 32;  W = woh; }
    else               { base = PTAR; ks = 8;  ldW = 32;  W = wtar; }
    int off   = i - base;
    int frag  = off >> 6;
    int r     = off & 63;
    int lane  = r >> 1;
    int j     = r & 1;
    int tile  = frag / ks;
    int kstep = frag % ks;
    int row   = tile * 16 + (lane & 15);
    int col   = 4 * kstep + 2 * (lane >> 4) + j;
    ws[i] = W[(size_t)row * ldW + col];
}

// ---------------------------------------------------------------------------
__global__ __launch_bounds__(256, 1)
void alpha_lstm_persistent(
    const float* __restrict__ feats,
    const float* __restrict__ pre_h, const float* __restrict__ pre_c,
    const float* __restrict__ out_h0, const float* __restrict__ out_c0,
    const float* __restrict__ pW0x, const float* __restrict__ pW0h,
    const float* __restrict__ pB0i, const float* __restrict__ pB0h,
    const float* __restrict__ pW1x, const float* __restrict__ pW1h,
    const float* __restrict__ pB1i, const float* __restrict__ pB1h,
    const float* __restrict__ fcW,  const float* __restrict__ fcB,
    const float* __restrict__ cW0x, const float* __restrict__ cW0h,
    const float* __restrict__ cB0i, const float* __restrict__ cB0h,
    const float* __restrict__ cW1x, const float* __restrict__ cW1h,
    const float* __restrict__ cB1i, const float* __restrict__ cB1h,
    const float* __restrict__ oWx,  const float* __restrict__ oWh,
    const float* __restrict__ oBi,  const float* __restrict__ oBh,
    const float* __restrict__ tW,   const float* __restrict__ tB,
    const float* __restrict__ dW,   const float* __restrict__ dB,
    const float* __restrict__ pk, float* __restrict__ out, int nsteps) {

    extern __shared__ float lds[];
    const int tid  = threadIdx.x;
    const int wave = tid >> 5;
    const int lane = tid & 31;

    // ---- one-time init: comm weights to padded LDS, recurrent state -------
    for (int i = tid; i < 128 * 64; i += 256) lds[O_W0  + (i >> 6) * 65 + (i & 63)] = cW0x[i];
    for (int i = tid; i < 128 * 32; i += 256) lds[O_WH0 + (i >> 5) * 33 + (i & 31)] = cW0h[i];
    for (int i = tid; i < 128 * 32; i += 256) lds[O_W1  + (i >> 5) * 33 + (i & 31)] = cW1x[i];
    for (int i = tid; i < 128 * 32; i += 256) lds[O_WH1 + (i >> 5) * 33 + (i & 31)] = cW1h[i];
    if (tid < 128) {
        lds[O_B0 + tid] = cB0i[tid] + cB0h[tid];
        lds[O_B1 + tid] = cB1i[tid] + cB1h[tid];
    }
    for (int i = tid; i < E_ * PRE_; i += 256) {
        int e = i >> 6, u = i & 63;
        lds[O_HP0 + e * LDH + u] = pre_h[e * 128 + u];
        lds[O_HP1 + e * LDH + u] = pre_h[e * 128 + 64 + u];
        lds[O_CP0 + e * LDH + u] = pre_c[e * 128 + u];
        lds[O_CP1 + e * LDH + u] = pre_c[e * 128 + 64 + u];
    }
    for (int i = tid; i < E_ * OUT_; i += 256) {
        int e = i >> 5, u = i & 31;
        lds[O_OH + e * LDOH + u] = out_h0[i];
        lds[O_OC + e * LDOH + u] = out_c0[i];
    }
#ifdef HAVE_TDM
    if (wave == 0 && nsteps > 0)                 // kick off prefetch of step 0
        tdm_load_feat(feats, (unsigned)(O_FEAT0 * sizeof(float)));
#endif
    __syncthreads();

    for (int t = 0; t < nsteps; ++t) {
        const int fbuf = (t & 1) ? O_FEAT1 : O_FEAT0;

#ifdef HAVE_TDM
        // Retire the async feature DMA for step t (wave 0 owns TENSORcnt),
        // publish to all waves, then immediately queue step t+1 into the
        // other buffer (its previous readers finished last iteration).
        if (wave == 0) __builtin_amdgcn_s_wait_tensorcnt(0);
        __syncthreads();
        if (wave == 0 && t + 1 < nsteps) {
            const int nbuf = ((t + 1) & 1) ? O_FEAT1 : O_FEAT0;
            tdm_load_feat(feats + (size_t)(t + 1) * (E_ * F_),
                          (unsigned)(nbuf * sizeof(float)));
        }
#else
        const float* fsrc = feats + (size_t)t * (E_ * F_);
        for (int c = tid; c < 512; c += 256) {
            int e = c >> 5, k4 = c & 31;
            *(v4f*)&lds[fbuf + e * LDF + k4 * 4] = *(const v4f*)(fsrc + c * 4);
        }
        __syncthreads();
#endif

        // ---- P1: preprocess LSTM layer 0 gates [16,256] (WMMA) ------------
#pragma unroll
        for (int jj = 0; jj < 2; ++jj) {
            int j = wave * 2 + jj;
            v8f acc = bias8(pB0i, pB0h, j * 16, lane);
            acc = mm_frag<32>(acc, &lds[fbuf], LDF, pk ? pk + P0X + j * 32 * 64 : nullptr, pW0x, 128, j * 16, lane);
            acc = mm_frag<16>(acc, &lds[O_HP0],  LDH, pk ? pk + P0H + j * 16 * 64 : nullptr, pW0h, 64,  j * 16, lane);
            store_frag(&lds[O_G], j * 16, lane, acc);
        }
        __syncthreads();
        for (int q = tid; q < E_ * PRE_; q += 256) {          // cell 0 update
            int e = q >> 6, u = q & 63;
            float gi = lds[O_G + e * LDG + u],       gf = lds[O_G + e * LDG + 64 + u];
            float gg = lds[O_G + e * LDG + 128 + u], go = lds[O_G + e * LDG + 192 + u];
            float c = sigm(gf) * lds[O_CP0 + e * LDH + u] + sigm(gi) * tanh_f(gg);
            lds[O_CP0 + e * LDH + u] = c;
            lds[O_HP0 + e * LDH + u] = sigm(go) * tanh_f(c);
        }
        __syncthreads();

        // ---- P2: preprocess LSTM layer 1 ----------------------------------
#pragma unroll
        for (int jj = 0; jj < 2; ++jj) {
            int j = wave * 2 + jj;
            v8f acc = bias8(pB1i, pB1h, j * 16, lane);
            acc = mm_frag<16>(acc, &lds[O_HP0], LDH, pk ? pk + P1X + j * 16 * 64 : nullptr, pW1x, 64, j * 16, lane);
            acc = mm_frag<16>(acc, &lds[O_HP1], LDH, pk ? pk + P1H + j * 16 * 64 : nullptr, pW1h, 64, j * 16, lane);
            store_frag(&lds[O_G], j * 16, lane, acc);
        }
        __syncthreads();
        for (int q = tid; q < E_ * PRE_; q += 256) {          // cell 1 update
            int e = q >> 6, u = q & 63;
            float gi = lds[O_G + e * LDG + u],       gf = lds[O_G + e * LDG + 64 + u];
            float gg = lds[O_G + e * LDG + 128 + u], go = lds[O_G + e * LDG + 192 + u];
            float c = sigm(gf) * lds[O_CP1 + e * LDH + u] + sigm(gi) * tanh_f(gg);
            lds[O_CP1 + e * LDH + u] = c;
            lds[O_HP1 + e * LDH + u] = sigm(go) * tanh_f(c);
        }
        __syncthreads();

        // ---- P3: fc + leaky_relu (waves 0-3); waves 4-7 build out_in feat --
        if (wave < 4) {
            int j = wave;
            v8f acc = bias8(fcB, nullptr, j * 16, lane);
            acc = mm_frag<16>(acc, &lds[O_HP1], LDH, pk ? pk + PFC + j * 16 * 64 : nullptr, fcW, 64, j * 16, lane);
            int n = j * 16 + (lane & 15), mh = (lane >> 4) * 8;
#pragma unroll
            for (int r = 0; r < 8; ++r) {
                float v = acc[r];
                v = v > 0.0f ? v : 0.05f * v;
                lds[O_X + (mh + r) * LDX + n] = v;
            }
        } else {
            for (int i = tid - 128; i < E_ * F_; i += 128) {
                int e = i >> 7, k = i & 127;
                lds[O_OI + e * LDOI + 64 + k] = lds[fbuf + e * LDF + k];
            }
        }
        if (tid < 32) { lds[O_CH0 + tid] = 0.0f; lds[O_CH1 + tid] = 0.0f; }
        __syncthreads();

        // ---- P4: sequential comm scan (48 alpha-LSTM micro-steps) ----------
        float cc0 = 0.0f, cc1 = 0.0f;                         // live in wave 0
        for (int s = 0; s < 48; ++s) {
            int e = s & 15;
            float alpha = (s < 16) ? 1.0f : ((s < 32) ? 0.333f : 0.110889f);
            const float* xe = &lds[O_X + e * LDX];
            int r = tid & 127;
            if (tid < 128) {                                   // Wih0 . x
                const float* wr = &lds[O_W0 + r * 65];
                float s0 = 0.0f, s1 = 0.0f;
#pragma unroll
                for (int k = 0; k < 64; k += 2) { s0 += wr[k] * xe[k]; s1 += wr[k + 1] * xe[k + 1]; }
                lds[O_PA + r] = s0 + s1;
            } else {                                           // Whh0 . h0
                const float* wr = &lds[O_WH0 + r * 33];
                float s0 = 0.0f;
#pragma unroll
                for (int k = 0; k < 32; ++k) s0 += wr[k] * lds[O_CH0 + k];
                lds[O_PB + r] = s0;
            }
            __syncthreads();
            if (wave == 0) {
                int u = lane;
                float gi = lds[O_B0 + u]      + lds[O_PA + u]      + lds[O_PB + u];
                float gf = lds[O_B0 + 32 + u] + lds[O_PA + 32 + u] + lds[O_PB + 32 + u];
                float gg = lds[O_B0 + 64 + u] + lds[O_PA + 64 + u] + lds[O_PB + 64 + u];
                float go = lds[O_B0 + 96 + u] + lds[O_PA + 96 + u] + lds[O_PB + 96 + u];
                cc0 = sigm(gf) * cc0 + alpha * sigm(gi) * tanh_f(gg);
                lds[O_CH0 + u] = sigm(go) * tanh_f(cc0);
            }
            __syncthreads();
            if (tid < 128) {                                   // Wih1 . nh0
                const float* wr = &lds[O_W1 + r * 33];
                float s0 = 0.0f;
#pragma unroll
                for (int k = 0; k < 32; ++k) s0 += wr[k] * lds[O_CH0 + k];
                lds[O_PA + r] = s0;
            } else {                                           // Whh1 . h1
                const float* wr = &lds[O_WH1 + r * 33];
                float s0 = 0.0f;
#pragma unroll
                for (int k = 0; k < 32; ++k) s0 += wr[k] * lds[O_CH1 + k];
                lds[O_PB + r] = s0;
            }
            __syncthreads();
            if (wave == 0) {
                int u = lane;
                float gi = lds[O_B1 + u]      + lds[O_PA + u]      + lds[O_PB + u];
                float gf = lds[O_B1 + 32 + u] + lds[O_PA + 32 + u] + lds[O_PB + 32 + u];
                float gg = lds[O_B1 + 64 + u] + lds[O_PA + 64 + u] + lds[O_PB + 64 + u];
                float go = lds[O_B1 + 96 + u] + lds[O_PA + 96 + u] + lds[O_PB + 96 + u];
                cc1 = sigm(gf) * cc1 + alpha * sigm(gi) * tanh_f(gg);
                lds[O_CH1 + u] = sigm(go) * tanh_f(cc1);
            }
            __syncthreads();
        }
        if (wave == 0) { lds[O_GRP + lane] = cc0; lds[O_GRP + 32 + lane] = cc1; }
        __syncthreads();
        for (int i = tid; i < E_ * 64; i += 256) {             // out_in group part
            int e = i >> 6, k = i & 63;
            lds[O_OI + e * LDOI + k] = lds[O_GRP + k];
        }
        __syncthreads();

        // ---- P5: output LSTM cell gates [16,128] (WMMA) --------------------
        {
            int j = wave;
            v8f acc = bias8(oBi, oBh, j * 16, lane);
            acc = mm_frag<48>(acc, &lds[O_OI], LDOI, pk ? pk + POW + j * 48 * 64 : nullptr, oWx, 192, j * 16, lane);
            acc = mm_frag<8>(acc, &lds[O_OH], LDOH, pk ? pk + POH + j * 8 * 64 : nullptr, oWh, 32, j * 16, lane);
            store_frag(&lds[O_G], j * 16, lane, acc);
        }
        __syncthreads();
        for (int q = tid; q < E_ * OUT_; q += 256) {
            int e = q >> 5, u = q & 31;
            float gi = lds[O_G + e * LDG + u],      gf = lds[O_G + e * LDG + 32 + u];
            float gg = lds[O_G + e * LDG + 64 + u], go = lds[O_G + e * LDG + 96 + u];
            float c = sigm(gf) * lds[O_OC + e * LDOH + u] + sigm(gi) * tanh_f(gg);
            lds[O_OC + e * LDOH + u] = c;
            lds[O_OH + e * LDOH + u] = sigm(go) * tanh_f(c);
        }
        __syncthreads();

        // ---- P6: heads: tar logits (WMMA, waves 0-3), dir logits -----------
        if (wave < 4) {
            int j = wave;
            v8f acc = bias8(tB, nullptr, j * 16, lane);
            acc = mm_frag<8>(acc, &lds[O_OH], LDOH, pk ? pk + PTAR + j * 8 * 64 : nullptr, tW, 32, j * 16, lane);
            store_frag(&lds[O_G], j * 16, lane, acc);
        } else if (tid >= 128 && tid < 176) {
            int idx = tid - 128, e = idx / 3, d = idx % 3;
            float s0 = dB[d];
            const float* wr = &dW[d * 32];
#pragma unroll
            for (int k = 0; k < 32; ++k) s0 += wr[k] * lds[O_OH + e * LDOH + k];
            lds[O_DIR + e * 3 + d] = s0;
        }
        __syncthreads();

        // ---- P7: softmax + global write ------------------------------------
        float* orow = out + (size_t)t * (E_ * (NF_ + 3));
        if (tid < 16) {
            int e = tid;
            float mx = -1e30f;
            for (int k = 0; k < NF_; ++k) mx = fmaxf(mx, lds[O_G + e * LDG + k]);
            float sum = 0.0f;
            for (int k = 0; k < NF_; ++k) sum += __expf(lds[O_G + e * LDG + k] - mx);
            float inv = 1.0f / sum;
            for (int k = 0; k < NF_; ++k)
                orow[e * 67 + k] = __expf(lds[O_G + e * LDG + k] - mx) * inv;
        } else if (tid < 32) {
            int e = tid - 16;
            float a = lds[O_DIR + e * 3], b = lds[O_DIR + e * 3 + 1], c = lds[O_DIR + e * 3 + 2];
            float mx = fmaxf(a, fmaxf(b, c));
            float ea = __expf(a - mx), eb = __expf(b - mx), ec = __expf(c - mx);
            float inv = 1.0f / (ea + eb + ec);
            orow[e * 67 + 64] = ea * inv;
            orow[e * 67 + 65] = eb * inv;
            orow[e * 67 + 66] = ec * inv;
        }
        __syncthreads();
    }
}

// ---------------------------------------------------------------------------
extern "C" void kernel_launch(void* const* d_in, const int* in_sizes, int n_in,
                              void* d_out, int out_size, void* d_ws, size_t ws_size,
                              hipStream_t stream) {
    const float* feats  = (const float*)d_in[0];
    const float* pre_h  = (const float*)d_in[1];
    const float* pre_c  = (const float*)d_in[2];
    const float* out_h0 = (const float*)d_in[3];
    const float* out_c0 = (const float*)d_in[4];
    const float* pW0x = (const float*)d_in[5];
    const float* pW0h = (const float*)d_in[6];
    const float* pB0i = (const float*)d_in[7];
    const float* pB0h = (const float*)d_in[8];
    const float* pW1x = (const float*)d_in[9];
    const float* pW1h = (const float*)d_in[10];
    const float* pB1i = (const float*)d_in[11];
    const float* pB1h = (const float*)d_in[12];
    const float* fcW  = (const float*)d_in[13];
    const float* fcB  = (const float*)d_in[14];
    const float* cW0x = (const float*)d_in[15];
    const float* cW0h = (const float*)d_in[16];
    const float* cB0i = (const float*)d_in[17];
    const float* cB0h = (const float*)d_in[18];
    const float* cW1x = (const float*)d_in[19];
    const float* cW1h = (const float*)d_in[20];
    const float* cB1i = (const float*)d_in[21];
    const float* cB1h = (const float*)d_in[22];
    const float* oWx  = (const float*)d_in[23];
    const float* oWh  = (const float*)d_in[24];
    const float* oBi  = (const float*)d_in[25];
    const float* oBh  = (const float*)d_in[26];
    const float* tW   = (const float*)d_in[27];
    const float* tB   = (const float*)d_in[28];
    const float* dW   = (const float*)d_in[29];
    const float* dB   = (const float*)d_in[30];

    const int nsteps = in_sizes[0] / (E_ * F_);

    const bool usePack = ws_size >= (size_t)PACK_TOTAL * sizeof(float);
    const float* pk = nullptr;
    if (usePack) {
        pack_kernel<<<(PACK_TOTAL + 255) / 256, 256, 0, stream>>>(
            (float*)d_ws, pW0x, pW0h, pW1x, pW1h, fcW, oWx, oWh, tW);
        pk = (const float*)d_ws;
    }

    const size_t shmem = (size_t)LDS_FLOATS * sizeof(float);
    alpha_lstm_persistent<<<1, 256, shmem, stream>>>(
        feats, pre_h, pre_c, out_h0, out_c0,
        pW0x, pW0h, pB0i, pB0h, pW1x, pW1h, pB1i, pB1h,
        fcW, fcB,
        cW0x, cW0h, cB0i, cB0h, cW1x, cW1h, cB1i, cB1h,
        oWx, oWh, oBi, oBh, tW, tB, dW, dB,
        pk, (float*)d_out, nsteps);
}